// AdaptiveDensityAwareSampler_81939386073384
// MI455X (gfx1250) — compile-verified
//
#include <hip/hip_runtime.h>
#include <math.h>

#define N_PTS            8192
#define ROWS_PER_BLOCK   256
#define CHUNK            1024
#define KKEEP            17          // ranks 0..16 (self at rank 0, r16 at rank 16)
#define TARGETP          1024
#define EPS_F            1e-8f
#define STG_STRIDE       20          // padded row stride (floats); 80B keeps 16B alignment

typedef float v2f __attribute__((ext_vector_type(2)));
typedef float v8f __attribute__((ext_vector_type(8)));

// ---------------- threefry2x32 (matches jax.random; key(42) => (0,42)) -------------
__device__ __forceinline__ unsigned rotl32(unsigned v, int r) {
    return (v << r) | (v >> (32 - r));
}

__device__ __forceinline__ void threefry2x32(unsigned k0, unsigned k1,
                                             unsigned x0, unsigned x1,
                                             unsigned& o0, unsigned& o1) {
    const unsigned ks2 = 0x1BD11BDAu ^ k0 ^ k1;
    unsigned X0 = x0 + k0, X1 = x1 + k1;
    const int ra[4] = {13, 15, 26, 6};
    const int rb[4] = {17, 29, 16, 24};
#pragma unroll
    for (int i = 0; i < 4; i++) { X0 += X1; X1 = rotl32(X1, ra[i]); X1 ^= X0; }
    X0 += k1;  X1 += ks2 + 1u;
#pragma unroll
    for (int i = 0; i < 4; i++) { X0 += X1; X1 = rotl32(X1, rb[i]); X1 ^= X0; }
    X0 += ks2; X1 += k0 + 2u;
#pragma unroll
    for (int i = 0; i < 4; i++) { X0 += X1; X1 = rotl32(X1, ra[i]); X1 ^= X0; }
    X0 += k0;  X1 += k1 + 3u;
#pragma unroll
    for (int i = 0; i < 4; i++) { X0 += X1; X1 = rotl32(X1, rb[i]); X1 ^= X0; }
    X0 += k1;  X1 += ks2 + 4u;
#pragma unroll
    for (int i = 0; i < 4; i++) { X0 += X1; X1 = rotl32(X1, ra[i]); X1 ^= X0; }
    X0 += ks2; X1 += k0 + 5u;
    o0 = X0; o1 = X1;
}

__device__ __forceinline__ float gumbel_noise(unsigned m) {
    // jax random_bits over 32768 elems: counter iota split into halves
    const unsigned half = 16384u;
    const unsigned sec  = (m >= half) ? 1u : 0u;
    const unsigned i0   = sec ? (m - half) : m;
    unsigned o0, o1;
    threefry2x32(0u, 42u, i0, i0 + half, o0, o1);
    const unsigned bits = sec ? o1 : o0;
    float u = __uint_as_float((bits >> 9) | 0x3f800000u) - 1.0f;   // [0,1)
    const float tiny = 1.1754943508222875e-38f;
    u = u * (1.0f - tiny) + tiny;
    u = fmaxf(u, tiny);
    return -logf(-logf(u));
}

// ---------------- kernel 1: WMMA-tiled kNN + density/PCA/score --------------------
__global__ __launch_bounds__(256) void knn_score_kernel(
    const float* __restrict__ pts,
    const float* __restrict__ alpha_p,
    const float* __restrict__ beta_p,
    const float* __restrict__ gamma_p,
    const float* __restrict__ cw,
    unsigned long long* __restrict__ keys) {
    __shared__ float4 s_chunk[CHUNK];                 // candidate points (x,y,z,|p|^2)
    __shared__ float  s_stage[8][32 * STG_STRIDE];    // per-wave 32x16 d^2 tile

    const int tid  = threadIdx.x;
    const int wave = tid >> 5;
    const int lane = tid & 31;
    const int lo16 = lane & 15;
    const int hi   = lane >> 4;

    const int blocks_per_batch = N_PTS / ROWS_PER_BLOCK;   // 32
    const int b  = blockIdx.x / blocks_per_batch;
    const int rb = blockIdx.x % blocks_per_batch;

    const int    row       = rb * ROWS_PER_BLOCK + wave * 32 + lane;  // owned point
    const size_t batch_off = (size_t)b * N_PTS;

    // ---- Build two A tiles (16x4 f32, K = x,y,z,|p|^2), exact WMMA A layout ----
    // VGPR0: lanes0-15 K=0 (x), lanes16-31 K=2 (z)
    // VGPR1: lanes0-15 K=1 (y), lanes16-31 K=3 (|p|^2)
    v2f a_t[2];
#pragma unroll
    for (int t = 0; t < 2; t++) {
        const int r = rb * ROWS_PER_BLOCK + wave * 32 + t * 16 + lo16;
        const float* p = pts + (batch_off + r) * 3;
        const float x = p[0], y = p[1], z = p[2];
        const float nrm = x * x + y * y + z * z;
        a_t[t][0] = hi ? z   : x;
        a_t[t][1] = hi ? nrm : y;
    }

    // ---- per-lane top-KKEEP over squared distances (replace-max, registers) ----
    float bd[KKEEP];
    int   bi[KKEEP];
#pragma unroll
    for (int s = 0; s < KKEEP; s++) { bd[s] = 3.4e38f; bi[s] = 0; }
    float curmax = 3.4e38f;
    int   maxpos = 0;

    float* stg = &s_stage[wave][0];

    for (int chunk = 0; chunk < N_PTS / CHUNK; ++chunk) {
        __syncthreads();   // protect previous chunk readers
        for (int i = tid; i < CHUNK; i += 256) {
            const float* p = pts + (batch_off + chunk * CHUNK + i) * 3;
            const float x = p[0], y = p[1], z = p[2];
            s_chunk[i] = make_float4(x, y, z, x * x + y * y + z * z);
            // prefetch next chunk's cache lines (global_prefetch_b8)
            if (chunk + 1 < N_PTS / CHUNK)
                __builtin_prefetch(pts + (batch_off + (chunk + 1) * CHUNK + i) * 3, 0, 1);
        }
        __syncthreads();

        for (int ct = 0; ct < CHUNK / 16; ++ct) {
            // B tile (4x16): VGPR0 lanes0-15 K=0 (-2x), lanes16-31 K=2 (-2z)
            //                VGPR1 lanes0-15 K=1 (-2y), lanes16-31 K=3 (1.0)
            const float4 pj = s_chunk[ct * 16 + lo16];
            v2f bv;
            bv[0] = hi ? (-2.0f * pj.z) : (-2.0f * pj.x);
            bv[1] = hi ? 1.0f           : (-2.0f * pj.y);
            const float cn = pj.w;  // |p_j|^2, per-lane (column N = lane%16)

#pragma unroll
            for (int t = 0; t < 2; t++) {
                v8f c = {};
                c = __builtin_amdgcn_wmma_f32_16x16x4_f32(
                        false, a_t[t], false, bv, (short)0, c, false, false);
#pragma unroll
                for (int v = 0; v < 8; ++v) {
                    const float d2   = c[v] + cn;             // + |p_j|^2 (squared dist)
                    const int   rloc = t * 16 + v + 8 * hi;   // C layout: M = v + 8*hi
                    stg[rloc * STG_STRIDE + lo16] = d2;       // sqrt deferred to epilogue
                }
            }

            // each lane scans its own row (local row == lane) of the 32x16 tile,
            // fetched as 4 x b128 to amortize DS latency/waits
            const int    jbase = chunk * CHUNK + ct * 16;
            const float* rowp  = &stg[lane * STG_STRIDE];
#pragma unroll 1
            for (int qd = 0; qd < 4; ++qd) {
                const float4 dq = *(const float4*)(rowp + qd * 4);
                const float dv[4] = {dq.x, dq.y, dq.z, dq.w};
#pragma unroll
                for (int cc = 0; cc < 4; ++cc) {
                    const float d2 = dv[cc];
                    const int   j  = jbase + qd * 4 + cc;
                    if (d2 < curmax) {
#pragma unroll
                        for (int s = 0; s < KKEEP; s++)
                            if (s == maxpos) { bd[s] = d2; bi[s] = j; }
                        curmax = -1.0f;
#pragma unroll
                        for (int s = 0; s < KKEEP; s++)
                            if (bd[s] > curmax) { curmax = bd[s]; maxpos = s; }
                    }
                }
            }
        }
    }

    // ---- extract sorted ranks 0..16 (by squared distance; ranking unchanged) ----
    int   idx16[16];
    float r16sq = 0.0f;
#pragma unroll
    for (int rank = 0; rank < KKEEP; ++rank) {
        float mn = 3.4e38f; int mpos = 0; int mid = 0;
#pragma unroll
        for (int s = 0; s < KKEEP; s++)
            if (bd[s] < mn) { mn = bd[s]; mpos = s; mid = bi[s]; }
        if (rank < 16) idx16[rank] = mid; else r16sq = mn;
#pragma unroll
        for (int s = 0; s < KKEEP; s++)
            if (s == mpos) bd[s] = 3.4e38f;
    }
    const float r16d = sqrtf(fmaxf(r16sq, 0.0f));   // the only sqrt per point

    // ---- PCA over 16 nearest (incl. self): cov = E[pp^T] - mm^T ----
    float sx = 0, sy = 0, sz = 0;
    float sxx = 0, sxy = 0, sxz = 0, syy = 0, syz = 0, szz = 0;
#pragma unroll
    for (int k = 0; k < 16; k++) {
        const float* p = pts + (batch_off + idx16[k]) * 3;
        const float x = p[0], y = p[1], z = p[2];
        sx += x; sy += y; sz += z;
        sxx += x * x; sxy += x * y; sxz += x * z;
        syy += y * y; syz += y * z; szz += z * z;
    }
    const float inv16 = 1.0f / 16.0f;
    const float mx = sx * inv16, my = sy * inv16, mz = sz * inv16;
    const float a00 = sxx * inv16 - mx * mx, a01 = sxy * inv16 - mx * my;
    const float a02 = sxz * inv16 - mx * mz, a11 = syy * inv16 - my * my;
    const float a12 = syz * inv16 - my * mz, a22 = szz * inv16 - mz * mz;

    // closed-form eigenvalues of symmetric 3x3
    const float q  = (a00 + a11 + a22) * (1.0f / 3.0f);
    const float p1 = a01 * a01 + a02 * a02 + a12 * a12;
    const float p2 = (a00 - q) * (a00 - q) + (a11 - q) * (a11 - q) +
                     (a22 - q) * (a22 - q) + 2.0f * p1;
    const float pp = sqrtf(fmaxf(p2, 0.0f) * (1.0f / 6.0f));
    float l1, l2, l3;
    if (pp < 1e-12f) {
        l1 = l2 = l3 = q;
    } else {
        const float ip  = 1.0f / pp;
        const float b00 = (a00 - q) * ip, b11 = (a11 - q) * ip, b22 = (a22 - q) * ip;
        const float b01 = a01 * ip, b02 = a02 * ip, b12 = a12 * ip;
        const float detB = b00 * (b11 * b22 - b12 * b12)
                         - b01 * (b01 * b22 - b12 * b02)
                         + b02 * (b01 * b12 - b11 * b02);
        const float r   = fminf(fmaxf(detB * 0.5f, -1.0f), 1.0f);
        const float phi = acosf(r) * (1.0f / 3.0f);
        l1 = q + 2.0f * pp * cosf(phi);
        l3 = q + 2.0f * pp * cosf(phi + 2.0943951023931953f);
        l2 = 3.0f * q - l1 - l3;
    }
    l1 = fabsf(l1); l2 = fabsf(l2); l3 = fabsf(l3);
    float tsw;
    if (l1 < l2) { tsw = l1; l1 = l2; l2 = tsw; }
    if (l2 < l3) { tsw = l2; l2 = l3; l3 = tsw; }
    if (l1 < l2) { tsw = l1; l1 = l2; l2 = tsw; }
    l1 += EPS_F; l2 += EPS_F; l3 += EPS_F;

    const float w0 = cw[0], w1 = cw[1], w2 = cw[2];
    const float il1 = 1.0f / l1;
    const float complexity = w0 * (l2 - l3) * il1 + w1 * (l1 - l2) * il1 + w2 * l3 * il1;

    // density at k=16, log-density, sigmoid
    const float r3      = r16d * r16d * r16d;
    const float density = 16.0f / (4.1887902047863905f * (r3 + EPS_F));
    const float logd    = logf(density + EPS_F);
    const float logit   = alpha_p[0] * logd + beta_p[0] * complexity + gamma_p[0];
    const float prob    = 1.0f / (1.0f + expf(-logit));

    // gumbel perturbed log-prob score
    const float g     = gumbel_noise((unsigned)(b * N_PTS + row));
    const float score = logf(prob + EPS_F) + g;

    // sortable key: flipped-float score in high 32 bits, index in low 32
    unsigned su = __float_as_uint(score);
    su = (su & 0x80000000u) ? ~su : (su | 0x80000000u);
    keys[(size_t)b * N_PTS + row] =
        ((unsigned long long)su << 32) | (unsigned)row;
}

// ---------------- kernel 2: per-batch bitonic top-1024 select --------------------
__global__ __launch_bounds__(1024) void sort_select_kernel(
    const float* __restrict__ pts,
    const unsigned long long* __restrict__ keys,
    float* __restrict__ out_pts,
    int* __restrict__ out_idx) {
    __shared__ unsigned long long sk[N_PTS];   // 64 KB

    const int b = blockIdx.x;
    const int t = threadIdx.x;

    for (int i = t; i < N_PTS; i += 1024)
        sk[i] = keys[(size_t)b * N_PTS + i];
    __syncthreads();

    // bitonic sort, descending
    for (int k = 2; k <= N_PTS; k <<= 1) {
        for (int j = k >> 1; j > 0; j >>= 1) {
            for (int i = t; i < N_PTS; i += 1024) {
                const int ixj = i ^ j;
                if (ixj > i) {
                    const bool up = ((i & k) == 0);   // descending run
                    const unsigned long long a = sk[i];
                    const unsigned long long c = sk[ixj];
                    const bool swp = up ? (a < c) : (a > c);
                    if (swp) { sk[i] = c; sk[ixj] = a; }
                }
            }
            __syncthreads();
        }
    }

    if (t < TARGETP) {
        const unsigned long long e = sk[t];
        const int idx = (int)(unsigned)(e & 0xffffffffu);
        const float* p = pts + ((size_t)b * N_PTS + idx) * 3;
        float* o = out_pts + ((size_t)b * TARGETP + t) * 3;
        o[0] = p[0]; o[1] = p[1]; o[2] = p[2];
        out_idx[b * TARGETP + t] = idx;
    }
}

extern "C" void kernel_launch(void* const* d_in, const int* in_sizes, int n_in,
                              void* d_out, int out_size, void* d_ws, size_t ws_size,
                              hipStream_t stream) {
    const float* pts     = (const float*)d_in[0];
    const float* alpha_p = (const float*)d_in[1];
    const float* beta_p  = (const float*)d_in[2];
    const float* gamma_p = (const float*)d_in[3];
    const float* cw      = (const float*)d_in[4];

    float* out_pts = (float*)d_out;                       // 4*1024*3 floats
    int*   out_idx = (int*)((float*)d_out + 4 * TARGETP * 3);

    unsigned long long* keys = (unsigned long long*)d_ws; // 4*8192*8 = 256 KB

    const int nblocks = 4 * (N_PTS / ROWS_PER_BLOCK);     // 128
    knn_score_kernel<<<nblocks, 256, 0, stream>>>(pts, alpha_p, beta_p, gamma_p, cw, keys);
    sort_select_kernel<<<4, 1024, 0, stream>>>(pts, keys, out_pts, out_idx);
}